// DivEncLayer_58317065945847
// MI455X (gfx1250) — compile-verified
//
#include <hip/hip_runtime.h>

typedef __attribute__((ext_vector_type(2))) float v2f;
typedef __attribute__((ext_vector_type(8))) float v8f;

#define BATCH 2048
#define Qn 128
#define Sn 128
#define Hn 32
#define Cn (Qn * Sn)

#define WAVES 8
#define RT 2                               // 16-row tiles per wave
#define ROWS_PER_BLOCK (WAVES * RT * 16)   // 256

__global__ __launch_bounds__(256)
void divenc_wmma_f32(const float* __restrict__ x,
                     const float* __restrict__ W1,
                     const float* __restrict__ b1,
                     const float* __restrict__ W2,
                     const float* __restrict__ b2,
                     float* __restrict__ out)
{
    // W1[q] staged in *fragment order*, 16 KB:
    //   frag f = k/4 (0..31), lane l (0..31): hi = l>>4, n = l&15
    //   w1s[f*128 + t*64 + l*2 + j] = W1[q][4f + 2*hi + j][n + 16*t]
    // -> B fragment for (f, tile t) is one aligned ds_load_b64, banks 2l/2l+1
    __shared__ float w1s[32 * 128];

    const int q   = blockIdx.x;
    const int tid = threadIdx.x;

    // ---- stage W1[q] into fragment-ordered LDS (2048 float pairs, 8/thread) ----
    {
        const float* w1g = W1 + (size_t)q * Sn * Hn;
        #pragma unroll
        for (int it = 0; it < 8; ++it) {
            int i  = tid + it * 256;       // pair index 0..2047
            int l  = i & 31;
            int t  = (i >> 5) & 1;
            int f  = i >> 6;
            int hi = l >> 4;
            int n  = l & 15;
            int s  = 4 * f + 2 * hi;
            int g  = s * Hn + n + 16 * t;
            float v0 = w1g[g];             // W1[s  ][n+16t]
            float v1 = w1g[g + Hn];        // W1[s+1][n+16t]
            float* dst = &w1s[f * 128 + t * 64 + l * 2];
            dst[0] = v0;
            dst[1] = v1;
        }
    }
    __syncthreads();

    const int wave = tid >> 5;
    const int lane = tid & 31;
    const int ln   = lane & 15;            // N / M index within half-wave
    const int hi   = lane >> 4;            // K sub-pair selector (0 or 1)
    const int b0   = blockIdx.y * ROWS_PER_BLOCK + wave * (RT * 16);

    // A-fragment row pointers: lane ln reads rows b0 + r*16 + ln, K offset 2*hi
    const float* xrow[RT];
    #pragma unroll
    for (int r = 0; r < RT; ++r)
        xrow[r] = x + (size_t)(b0 + r * 16 + ln) * Cn + q * Sn + hi * 2;

    const float* bfrag = &w1s[lane * 2];   // shared address reg; frag offset is immediate

    v8f acc[RT][2] = {};                   // 16x16 f32 accumulators (2 N-tiles of H)

    #pragma unroll 8
    for (int k = 0; k < Sn; k += 4) {
        const int f = k >> 2;
        v2f bfa = *(const v2f*)(bfrag + f * 128);        // tile 0 (h 0..15)
        v2f bfb = *(const v2f*)(bfrag + f * 128 + 64);   // tile 1 (h 16..31)

        #pragma unroll
        for (int r = 0; r < RT; ++r) {
            // x is single-use: non-temporal so the 128 MB stream doesn't evict W1
            v2f a = __builtin_nontemporal_load((const v2f*)(xrow[r] + k));
            acc[r][0] = __builtin_amdgcn_wmma_f32_16x16x4_f32(
                false, a, false, bfa, (short)0, acc[r][0], false, false);
            acc[r][1] = __builtin_amdgcn_wmma_f32_16x16x4_f32(
                false, a, false, bfb, (short)0, acc[r][1], false, false);
        }
    }

    // ---- epilogue: bias, ELU, dot with W2, half-wave reduction, store ----
    const float w2a = W2[q * Hn + ln];
    const float w2b = W2[q * Hn + ln + 16];
    const float b1a = b1[q * Hn + ln];
    const float b1b = b1[q * Hn + ln + 16];
    const float b2q = b2[q];

    #pragma unroll
    for (int r = 0; r < RT; ++r) {
        #pragma unroll
        for (int m = 0; m < 8; ++m) {
            float h0 = acc[r][0][m] + b1a;
            float h1 = acc[r][1][m] + b1b;
            h0 = (h0 > 0.0f) ? h0 : (__expf(h0) - 1.0f);   // ELU(alpha=1)
            h1 = (h1 > 0.0f) ? h1 : (__expf(h1) - 1.0f);
            float p = h0 * w2a + h1 * w2b;
            // reduce the 16 N-lanes inside each half-wave
            p += __shfl_xor(p, 8, 16);
            p += __shfl_xor(p, 4, 16);
            p += __shfl_xor(p, 2, 16);
            p += __shfl_xor(p, 1, 16);
            if (ln == 0) {
                // D layout: VGPR m -> row m (lanes 0-15) / row m+8 (lanes 16-31)
                int b = b0 + r * 16 + hi * 8 + m;
                out[(size_t)b * Qn + q] = p + b2q;
            }
        }
    }
}

extern "C" void kernel_launch(void* const* d_in, const int* in_sizes, int n_in,
                              void* d_out, int out_size, void* d_ws, size_t ws_size,
                              hipStream_t stream) {
    const float* x  = (const float*)d_in[0];
    const float* W1 = (const float*)d_in[1];
    const float* b1 = (const float*)d_in[2];
    const float* W2 = (const float*)d_in[3];
    const float* b2 = (const float*)d_in[4];
    float* out = (float*)d_out;

    dim3 grid(Qn, BATCH / ROWS_PER_BLOCK);     // (128, 8)
    divenc_wmma_f32<<<grid, 256, 0, stream>>>(x, W1, b1, W2, b2, out);
}